// PQLayer_80728205295910
// MI455X (gfx1250) — compile-verified
//
#include <hip/hip_runtime.h>
#include <hip/hip_bf16.h>
#include <math.h>

typedef __attribute__((ext_vector_type(2))) float v2f;
typedef __attribute__((ext_vector_type(8))) float v8f;

#define B_N   8192
#define FEAT  1024
#define M_N   32
#define K_N   256
#define D_N   32

// d_out element offsets (tuple concatenated flat in return order)
static constexpr size_t OFF_XHAT  = 0;
static constexpr size_t OFF_HARD  = (size_t)B_N * FEAT;                  // 8388608
static constexpr size_t OFF_SOFT  = OFF_HARD + (size_t)B_N * M_N;        // 8650752
static constexpr size_t OFF_XNORM = OFF_SOFT + (size_t)B_N * M_N * K_N;  // 75759616
static constexpr size_t OFF_REG   = OFF_XNORM + (size_t)B_N * FEAT;      // 84148224

// LDS padded strides (floats) to avoid 64-bank conflicts on 128B-strided access
#define CS 34   // codebook row stride (32 + pad, even to keep v2f alignment)
#define XS 34   // x tile row stride
#define SS 258  // sims row stride

__device__ __forceinline__ float wred_sum(float v) {
  #pragma unroll
  for (int o = 16; o >= 1; o >>= 1) v += __shfl_xor(v, o, 32);
  return v;
}

// ---- init: zero per-m stats in ws and the reg_loss output region ----
__global__ void k_init(float* __restrict__ dout, float* __restrict__ ws) {
  int t = blockIdx.x * blockDim.x + threadIdx.x;
  if (t < 64) ws[t] = 0.0f;                 // ws[0..31]=sum, ws[32..63]=sumsq
  if (t < B_N) dout[OFF_REG + t] = 0.0f;
}

// ---- k0: per-(b,m) L2 normalize; one wave32 per 32-float group ----
__global__ __launch_bounds__(256) void k0_normalize(const float* __restrict__ x,
                                                    float* __restrict__ dout) {
  int gw   = (blockIdx.x * blockDim.x + threadIdx.x) >> 5;
  int lane = threadIdx.x & 31;
  if (gw >= B_N * M_N) return;
  int b = gw >> 5;
  int m = gw & 31;
  size_t idx = (size_t)b * FEAT + m * D_N + lane;
  float v = __builtin_nontemporal_load(&x[idx]);   // x read exactly once -> NT
  float s = wred_sum(v * v);
  float n = sqrtf(s);
  float d = fmaxf(n, 1e-12f);
  // x_normalized is re-read by k1/k3 -> keep regular temporal policy (L2-resident)
  dout[OFF_XNORM + idx] = v / d;
}

// ---- k1: WMMA GEMM, accumulate per-m sum & sumsq of sims (no sims stored) ----
__global__ __launch_bounds__(128) void k1_stats(const float* __restrict__ Ck,
                                                const float* __restrict__ xn,
                                                float* __restrict__ ws) {
  __shared__ float sC[K_N * CS];
  __shared__ float sX[16 * XS];
  const int m    = blockIdx.y;
  const int b0   = blockIdx.x * 16;
  const int t    = threadIdx.x;
  const int lane = t & 31, wave = t >> 5;
  const int half = lane >> 4, l16 = lane & 15;

  const float* Cg = Ck + (size_t)m * (K_N * D_N);
  for (int i2 = t; i2 < (K_N * D_N) / 2; i2 += 128) {       // codebook -> LDS (coalesced 8B)
    int k = i2 >> 4, d = (i2 & 15) * 2;
    *(v2f*)&sC[k * CS + d] = *(const v2f*)&Cg[i2 * 2];
  }
  for (int i2 = t; i2 < (16 * D_N) / 2; i2 += 128) {        // x_norm tile -> LDS
    int r = i2 >> 4, d = (i2 & 15) * 2;
    *(v2f*)&sX[r * XS + d] = *(const v2f*)&xn[(size_t)(b0 + r) * FEAT + m * D_N + d];
  }
  __syncthreads();

  v2f a[8];
  #pragma unroll
  for (int s = 0; s < 8; ++s)                               // A frag: M=l16, K=4s+2*half+{0,1}
    a[s] = *(const v2f*)&sX[l16 * XS + 4 * s + 2 * half];

  float lsum = 0.0f, lsq = 0.0f;
  #pragma unroll
  for (int tt = 0; tt < 4; ++tt) {
    int tile = wave * 4 + tt;                               // 16 codes per tile
    v8f acc = {0.f, 0.f, 0.f, 0.f, 0.f, 0.f, 0.f, 0.f};
    #pragma unroll
    for (int s = 0; s < 8; ++s) {
      v2f bf = *(const v2f*)&sC[(tile * 16 + l16) * CS + 4 * s + 2 * half];
      acc = __builtin_amdgcn_wmma_f32_16x16x4_f32(false, a[s], false, bf,
                                                  (short)0, acc, false, false);
    }
    #pragma unroll
    for (int j = 0; j < 8; ++j) { float v = acc[j]; lsum += v; lsq += v * v; }
  }
  lsum = wred_sum(lsum);
  lsq  = wred_sum(lsq);
  if (lane == 0) {
    atomicAdd(&ws[m], lsum);
    atomicAdd(&ws[32 + m], lsq);
  }
}

// ---- k2: finalize per-m mean / inv_std ----
__global__ void k2_finalize(float* __restrict__ ws) {
  int m = threadIdx.x;
  if (m >= M_N) return;
  const float cnt = (float)((size_t)B_N * K_N);
  float mean = ws[m] / cnt;
  float var  = ws[32 + m] / cnt - mean * mean;
  ws[64 + m] = mean;
  ws[96 + m] = rsqrtf(var + 1e-5f);
}

// ---- k3: WMMA GEMM + BN + argmax + softmax + gather + residual ----
__global__ __launch_bounds__(128) void k3_main(const float* __restrict__ Ck,
                                               float* __restrict__ dout,
                                               const float* __restrict__ ws) {
  __shared__ float sC[K_N * CS];   // 34816 B
  __shared__ float sX[16 * XS];    //  2176 B
  __shared__ float sS[16 * SS];    // 16512 B
  const int m    = blockIdx.y;
  const int b0   = blockIdx.x * 16;
  const int t    = threadIdx.x;
  const int lane = t & 31, wave = t >> 5;
  const int half = lane >> 4, l16 = lane & 15;

  const float* Cg = Ck + (size_t)m * (K_N * D_N);
  const float* xn = dout + OFF_XNORM;
  for (int i2 = t; i2 < (K_N * D_N) / 2; i2 += 128) {
    int k = i2 >> 4, d = (i2 & 15) * 2;
    *(v2f*)&sC[k * CS + d] = *(const v2f*)&Cg[i2 * 2];
  }
  for (int i2 = t; i2 < (16 * D_N) / 2; i2 += 128) {
    int r = i2 >> 4, d = (i2 & 15) * 2;
    *(v2f*)&sX[r * XS + d] = *(const v2f*)&xn[(size_t)(b0 + r) * FEAT + m * D_N + d];
  }
  __syncthreads();

  v2f a[8];
  #pragma unroll
  for (int s = 0; s < 8; ++s)
    a[s] = *(const v2f*)&sX[l16 * XS + 4 * s + 2 * half];

  #pragma unroll
  for (int tt = 0; tt < 4; ++tt) {
    int tile = wave * 4 + tt;
    v8f acc = {0.f, 0.f, 0.f, 0.f, 0.f, 0.f, 0.f, 0.f};
    #pragma unroll
    for (int s = 0; s < 8; ++s) {
      v2f bf = *(const v2f*)&sC[(tile * 16 + l16) * CS + 4 * s + 2 * half];
      acc = __builtin_amdgcn_wmma_f32_16x16x4_f32(false, a[s], false, bf,
                                                  (short)0, acc, false, false);
    }
    // C/D layout: VGPR j -> row j (lanes 0-15) / row j+8 (lanes 16-31), col = l16
    #pragma unroll
    for (int j = 0; j < 8; ++j)
      sS[(j + 8 * half) * SS + tile * 16 + l16] = acc[j];
  }
  __syncthreads();

  const float mean = ws[64 + m];
  const float istd = ws[96 + m];
  float* soft = dout + OFF_SOFT;
  int*   hard = (int*)dout + OFF_HARD;

  for (int row = wave; row < 16; row += 4) {   // 4 rows per wave
    int b = b0 + row;
    float v[8];
    float mx = -3.402823466e38f;
    int   ai = 0;
    #pragma unroll
    for (int j = 0; j < 8; ++j) {              // 8 codes per lane
      v[j] = sS[row * SS + lane + 32 * j];
      if (v[j] > mx) { mx = v[j]; ai = lane + 32 * j; }
    }
    #pragma unroll
    for (int o = 16; o >= 1; o >>= 1) {        // argmax, first-index tie-break
      float om = __shfl_xor(mx, o, 32);
      int   oi = __shfl_xor(ai, o, 32);
      if (om > mx || (om == mx && oi < ai)) { mx = om; ai = oi; }
    }
    // BN is monotone (istd>0), so argmax on raw sims == argmax on BN sims
    float bnmax = (mx - mean) * istd;
    float e[8], ls = 0.0f;
    #pragma unroll
    for (int j = 0; j < 8; ++j) {
      float bn = (v[j] - mean) * istd;
      e[j] = __expf(bn - bnmax);
      ls += e[j];
    }
    ls = wred_sum(ls);
    float inv = 1.0f / ls;
    size_t sbase = ((size_t)b * M_N + m) * K_N;
    // soft_codes: 268 MB streamed once, never re-read -> nontemporal stores
    // (keep L2 free for codebook + x_normalized re-reads)
    #pragma unroll
    for (int j = 0; j < 8; ++j)
      __builtin_nontemporal_store(e[j] * inv, &soft[sbase + lane + 32 * j]);
    if (lane == 0) hard[(size_t)b * M_N + m] = ai;

    // gather codeword (x_hat forward value == x_hat_hard) + residual loss
    float cd = sC[ai * CS + lane];
    float xd = sX[row * XS + lane];
    __builtin_nontemporal_store(cd, &dout[OFF_XHAT + (size_t)b * FEAT + m * D_N + lane]);
    float df = xd - cd;
    float ss = wred_sum(df * df);
    if (lane == 0) atomicAdd(&dout[OFF_REG + b], ss);
  }
}

extern "C" void kernel_launch(void* const* d_in, const int* in_sizes, int n_in,
                              void* d_out, int out_size, void* d_ws, size_t ws_size,
                              hipStream_t stream) {
  const float* x  = (const float*)d_in[0];   // (B, 1024) f32
  const float* Ck = (const float*)d_in[1];   // (32, 256, 32) f32
  float* out = (float*)d_out;
  float* ws  = (float*)d_ws;

  k_init<<<32, 256, 0, stream>>>(out, ws);
  k0_normalize<<<(B_N * M_N) / 8, 256, 0, stream>>>(x, out);
  dim3 g(B_N / 16, M_N);
  k1_stats<<<g, 128, 0, stream>>>(Ck, out + OFF_XNORM, ws);
  k2_finalize<<<1, 32, 0, stream>>>(ws);
  k3_main<<<g, 128, 0, stream>>>(Ck, out, ws);
}